// Sampler_19361712570998
// MI455X (gfx1250) — compile-verified
//
#include <hip/hip_runtime.h>
#include <hip/hip_bf16.h>
#include <math.h>

#define B_   64
#define S_   8
#define D_   2048
#define V_   128000
#define CAP  1024

typedef float v2f __attribute__((ext_vector_type(2)));
typedef float v8f __attribute__((ext_vector_type(8)));

// ---------------------------------------------------------------------------
// Kernel 1: logits[64,128000] = x[:, pos, :] @ emb^T + bias  via f32 WMMA.
// One wave computes a 64x16 tile (4 accumulators reuse one B fragment).
// emb is streamed with non-temporal loads (1.05 GB, HBM-bound); h hits L2.
// ---------------------------------------------------------------------------
__global__ __launch_bounds__(128, 1) void logits_gemm_wmma(
    const float* __restrict__ x, const float* __restrict__ emb,
    const float* __restrict__ ebias, const int* __restrict__ out_pos,
    float* __restrict__ logits)
{
    const int pos  = out_pos[0];
    const int wave = (int)(threadIdx.x >> 5);
    const int lane = (int)(threadIdx.x & 31);
    const int nc   = lane & 15;          // N within tile
    const int kh   = lane >> 4;          // K-half selector (0 or 1)
    const int n    = (int)blockIdx.x * 64 + wave * 16 + nc;

    // B fragment (4x16 of emb^T): lane loads emb[n][k + 2*kh .. +1] as float2
    const float* bptr  = emb + (size_t)n * D_ + (size_t)(kh << 1);
    // A fragments (16x4 of h): lane loads h[m][k + 2*kh .. +1] as float2
    const float* abase = x + ((size_t)nc * S_ + (size_t)pos) * D_ + (size_t)(kh << 1);
    const size_t MT = (size_t)16 * S_ * D_;   // stride between M-tiles of 16 rows

    v8f acc0 = {}; v8f acc1 = {}; v8f acc2 = {}; v8f acc3 = {};

#pragma unroll 4
    for (int k = 0; k < D_; k += 4) {
        v2f bf = __builtin_nontemporal_load((const v2f*)(bptr + k));
        v2f a0 = *(const v2f*)(abase + k);
        v2f a1 = *(const v2f*)(abase + MT + k);
        v2f a2 = *(const v2f*)(abase + 2 * MT + k);
        v2f a3 = *(const v2f*)(abase + 3 * MT + k);
        acc0 = __builtin_amdgcn_wmma_f32_16x16x4_f32(false, a0, false, bf, (short)0, acc0, false, false);
        acc1 = __builtin_amdgcn_wmma_f32_16x16x4_f32(false, a1, false, bf, (short)0, acc1, false, false);
        acc2 = __builtin_amdgcn_wmma_f32_16x16x4_f32(false, a2, false, bf, (short)0, acc2, false, false);
        acc3 = __builtin_amdgcn_wmma_f32_16x16x4_f32(false, a3, false, bf, (short)0, acc3, false, false);
    }

    // C/D layout: VGPR r holds M = r (lanes 0-15) or M = 8+r (lanes 16-31)
    const float bv   = ebias[n];
    const int   mloc = kh << 3;
#pragma unroll
    for (int r = 0; r < 8; ++r) {
        logits[(size_t)(mloc + r)      * V_ + n] = acc0[r] + bv;
        logits[(size_t)(mloc + r + 16) * V_ + n] = acc1[r] + bv;
        logits[(size_t)(mloc + r + 32) * V_ + n] = acc2[r] + bv;
        logits[(size_t)(mloc + r + 48) * V_ + n] = acc3[r] + bv;
    }
}

// ---------------------------------------------------------------------------
// Softmax statistics per row: max and sum(exp(l - max)). Logits are L2-hot.
// ---------------------------------------------------------------------------
__global__ __launch_bounds__(256) void row_stats_kernel(
    const float* __restrict__ logits, float* __restrict__ stats)
{
    const int b = (int)blockIdx.x;
    const int tid = (int)threadIdx.x;
    const float* row = logits + (size_t)b * V_;
    __shared__ float red[256];

    float m = -INFINITY;
    for (int i = tid; i < V_; i += 256) m = fmaxf(m, row[i]);
    red[tid] = m; __syncthreads();
    for (int s = 128; s > 0; s >>= 1) {
        if (tid < s) red[tid] = fmaxf(red[tid], red[tid + s]);
        __syncthreads();
    }
    m = red[0]; __syncthreads();

    float sum = 0.f;
    for (int i = tid; i < V_; i += 256) sum += __expf(row[i] - m);
    red[tid] = sum; __syncthreads();
    for (int s = 128; s > 0; s >>= 1) {
        if (tid < s) red[tid] += red[tid + s];
        __syncthreads();
    }
    if (tid == 0) { stats[2 * b] = m; stats[2 * b + 1] = red[0]; }
}

__device__ __forceinline__ unsigned ordkey(float f) {
    unsigned u = __float_as_uint(f);
    return (u & 0x80000000u) ? ~u : (u | 0x80000000u);  // monotone: bigger key = bigger float
}

// ---------------------------------------------------------------------------
// Top-byte histogram -> threshold bin that contains the 64th largest logit.
// Also resets the per-row candidate counter (determinism across replays).
// ---------------------------------------------------------------------------
__global__ __launch_bounds__(256) void topk_hist_kernel(
    const float* __restrict__ logits, int* __restrict__ thr)
{
    const int b = (int)blockIdx.x;
    const int tid = (int)threadIdx.x;
    __shared__ int hist[256];
    hist[tid] = 0; __syncthreads();

    const float* row = logits + (size_t)b * V_;
    for (int i = tid; i < V_; i += 256)
        atomicAdd(&hist[ordkey(row[i]) >> 24], 1);
    __syncthreads();

    if (tid == 0) {
        int acc = 0, bin = 0;
        for (int j = 255; j >= 0; --j) {
            if (acc + hist[j] >= 64 || j == 0) { bin = j; break; }
            acc += hist[j];
        }
        thr[2 * b] = bin;
        thr[2 * b + 1] = 0;   // candidate counter for the gather kernel
    }
}

// ---------------------------------------------------------------------------
// Gather all logits whose top key-byte >= threshold bin (superset of top-64).
// ---------------------------------------------------------------------------
__global__ __launch_bounds__(256) void topk_gather_kernel(
    const float* __restrict__ logits, int* __restrict__ thr,
    float* __restrict__ candv, int* __restrict__ candi)
{
    const int b = (int)blockIdx.x;
    const int bin = thr[2 * b];
    const float* row = logits + (size_t)b * V_;
    for (int i = (int)threadIdx.x; i < V_; i += 256) {
        float v = row[i];
        if ((int)(ordkey(v) >> 24) >= bin) {
            int slot = atomicAdd(&thr[2 * b + 1], 1);
            if (slot < CAP) {
                candv[(size_t)b * CAP + slot] = v;
                candi[(size_t)b * CAP + slot] = i;
            }
        }
    }
}

// ---------------------------------------------------------------------------
// Per row: extract top-64 from candidates, apply top-p (exclusive-cumsum)
// and top-k masks, renormalize, draw categorical sample; write token id.
// ---------------------------------------------------------------------------
__global__ __launch_bounds__(256) void sample_kernel(
    const float* __restrict__ candv, const int* __restrict__ candi,
    const int* __restrict__ thr, const float* __restrict__ stats,
    const float* __restrict__ top_ps, const int* __restrict__ top_ks,
    float* __restrict__ out_tok)
{
    const int b = (int)blockIdx.x;
    const int tid = (int)threadIdx.x;
    const int n = min(thr[2 * b + 1], CAP);

    __shared__ float vals[CAP];
    __shared__ int   idxs[CAP];
    __shared__ float rv[256];
    __shared__ int   ri[256];
    __shared__ float sv[64];
    __shared__ int   si[64];
    __shared__ float pw[64];

    for (int i = tid; i < CAP; i += 256) {
        vals[i] = (i < n) ? candv[(size_t)b * CAP + i] : -INFINITY;
        idxs[i] = (i < n) ? candi[(size_t)b * CAP + i] : -1;
    }
    __syncthreads();

    // 64 rounds of parallel argmax extraction -> descending sorted top-64
    for (int r = 0; r < 64; ++r) {
        float best = -INFINITY; int bi = -1;
        for (int i = tid; i < CAP; i += 256)
            if (vals[i] > best) { best = vals[i]; bi = i; }
        rv[tid] = best; ri[tid] = bi; __syncthreads();
        for (int s = 128; s > 0; s >>= 1) {
            if (tid < s && rv[tid + s] > rv[tid]) { rv[tid] = rv[tid + s]; ri[tid] = ri[tid + s]; }
            __syncthreads();
        }
        if (tid == 0) {
            sv[r] = rv[0];
            si[r] = (ri[0] >= 0) ? idxs[ri[0]] : -1;
            if (ri[0] >= 0) vals[ri[0]] = -INFINITY;
        }
        __syncthreads();
    }

    if (tid == 0) {
        const float m = stats[2 * b], ssum = stats[2 * b + 1];
        const float p = top_ps[b];
        const int   kk = top_ks[b];
        float cum = 0.f, kept = 0.f;
        for (int r = 0; r < 64; ++r) {
            float prob = (si[r] >= 0 && sv[r] > -1e37f) ? __expf(sv[r] - m) / ssum : 0.f;
            bool drop = (cum > p) || (r >= kk);   // exclusive-cumsum top-p, then top-k
            cum += prob;
            pw[r] = drop ? 0.f : prob;
            kept += pw[r];
        }
        // hash-based uniform draw (jax key(42) categorical is not reproducible here)
        unsigned z = 0x2545F491u ^ (42u + 0x9E3779B9u * (unsigned)(b + 1));
        z ^= z >> 16; z *= 0x85EBCA6Bu; z ^= z >> 13; z *= 0xC2B2AE35u; z ^= z >> 16;
        const float u = (float)(z >> 8) * (1.0f / 16777216.0f);
        const float target = u * kept;
        int chosen = (si[0] >= 0) ? si[0] : 0;
        float acc = 0.f;
        for (int r = 0; r < 64; ++r) {
            if (pw[r] > 0.f) {
                acc += pw[r];
                if (acc >= target) { chosen = si[r]; break; }
            }
        }
        out_tok[b] = (float)chosen;
    }
}

extern "C" void kernel_launch(void* const* d_in, const int* in_sizes, int n_in,
                              void* d_out, int out_size, void* d_ws, size_t ws_size,
                              hipStream_t stream) {
    const float* x       = (const float*)d_in[0];   // [64, 8, 2048]
    const float* emb     = (const float*)d_in[1];   // [128000, 2048]
    const float* ebias   = (const float*)d_in[2];   // [128000]
    const float* top_ps  = (const float*)d_in[3];   // [64]
    const int*   out_pos = (const int*)d_in[4];     // [1]
    const int*   top_ks  = (const int*)d_in[5];     // [64]
    (void)in_sizes; (void)n_in; (void)out_size; (void)ws_size;

    float* out    = (float*)d_out;
    float* tokens = out;            // first output: 64 token ids
    float* logits = out + B_;       // second output: [64, 128000]

    // workspace layout
    float* stats = (float*)d_ws;                    // 2*64 floats
    int*   thr   = (int*)(stats + 2 * B_);          // 2*64 ints (bin, counter)
    float* candv = (float*)(thr + 2 * B_);          // 64*CAP floats
    int*   candi = (int*)(candv + (size_t)B_ * CAP);// 64*CAP ints

    logits_gemm_wmma<<<dim3(V_ / 64), dim3(128), 0, stream>>>(x, emb, ebias, out_pos, logits);
    row_stats_kernel<<<dim3(B_), dim3(256), 0, stream>>>(logits, stats);
    topk_hist_kernel<<<dim3(B_), dim3(256), 0, stream>>>(logits, thr);
    topk_gather_kernel<<<dim3(B_), dim3(256), 0, stream>>>(logits, thr, candv, candi);
    sample_kernel<<<dim3(B_), dim3(256), 0, stream>>>(candv, candi, thr, stats, top_ps, top_ks, tokens);
}